// GCN_31559419691025
// MI455X (gfx1250) — compile-verified
//
#include <hip/hip_runtime.h>

#define DINK 128
#define DHK  128
#define DOK  64

typedef __attribute__((ext_vector_type(2))) float v2f;
typedef __attribute__((ext_vector_type(8))) float v8f;

// ---------------- utility kernels ----------------

__global__ void gcn_zero_f4(float4* __restrict__ p, long n4) {
    long i = (long)blockIdx.x * blockDim.x + threadIdx.x;
    if (i < n4) p[i] = make_float4(0.f, 0.f, 0.f, 0.f);
}

__global__ void gcn_deg(const int* __restrict__ dst, float* __restrict__ deg, int E) {
    int e = blockIdx.x * blockDim.x + threadIdx.x;
    if (e < E) atomicAdd(&deg[dst[e]], 1.0f);
}

// in-place: deg -> deg^{-1/2} (0 where deg==0)
__global__ void gcn_dinv(float* __restrict__ deg, int n) {
    int i = blockIdx.x * blockDim.x + threadIdx.x;
    if (i < n) {
        float d = deg[i];
        deg[i] = (d > 0.f) ? rsqrtf(d) : 0.f;
    }
}

// ---------------- WMMA fp32 GEMM: out[M x (NT*16)] = H[M x 128] @ W^T ----------------
// W is row-major [NT*16, 128]; B[k][n] = W[n][k].
// One block = one 16-row M tile; wave w handles column tile w. NT waves per block.
template<int NT>
__global__ __launch_bounds__(NT * 32)
void gcn_gemm_wmma(const float* __restrict__ H, const float* __restrict__ W,
                   float* __restrict__ out) {
    const int lane  = threadIdx.x & 31;
    const int wave  = threadIdx.x >> 5;
    const int baseM = blockIdx.x << 4;
    const int l15   = lane & 15;
    const int half  = lane >> 4;          // 0 or 1
    const int col   = (wave << 4) + l15;  // output column (N)
    const int rowA  = baseM + l15;        // input row (M)
    const int koff  = half << 1;          // 0 or 2

    const float* hp = H + (size_t)rowA * DINK + koff;
    const float* wp = W + (size_t)col  * DINK + koff;

    v8f acc = {0.f, 0.f, 0.f, 0.f, 0.f, 0.f, 0.f, 0.f};

#pragma unroll
    for (int k0 = 0; k0 < DINK; k0 += 4) {
        v2f a = *(const v2f*)(hp + k0);   // (M=rowA, K=k0+koff), (K=k0+koff+1)
        v2f b = *(const v2f*)(wp + k0);   // (N=col , K=k0+koff), (K=k0+koff+1)
        acc = __builtin_amdgcn_wmma_f32_16x16x4_f32(
            /*neg_a=*/false, a, /*neg_b=*/false, b,
            /*c_mod=*/(short)0, acc, /*reuse_a=*/false, /*reuse_b=*/false);
    }

    // C/D layout: VGPR r, lanes 0-15 -> (M=r, N=lane); lanes 16-31 -> (M=8+r, N=lane-16)
    const int ldout = NT * 16;
    float* op = out + (size_t)(baseM + (half << 3)) * ldout + col;
#pragma unroll
    for (int r = 0; r < 8; ++r)
        op[(size_t)r * ldout] = acc[r];
}

// ---------------- edge scatter: out[dst] += tmp[src] * dinv[src]*dinv[dst] ----------------
// D=128: one wave per edge, lane handles 4 consecutive feats (coalesced).
__global__ void gcn_scatter128(const int* __restrict__ src, const int* __restrict__ dst,
                               const float* __restrict__ dinv, const float* __restrict__ tmp,
                               float* __restrict__ out, int E) {
    long tid = (long)blockIdx.x * blockDim.x + threadIdx.x;
    int e = (int)(tid >> 5);
    if (e >= E) return;
    int lane = (int)(tid & 31);
    int s = src[e], d = dst[e];
    float nrm = dinv[s] * dinv[d];
    const float4 v = *(const float4*)(tmp + (size_t)s * 128 + lane * 4);
    float* o = out + (size_t)d * 128 + lane * 4;
    atomicAdd(o + 0, v.x * nrm);
    atomicAdd(o + 1, v.y * nrm);
    atomicAdd(o + 2, v.z * nrm);
    atomicAdd(o + 3, v.w * nrm);
}

// D=64: 16 lanes per edge (2 edges per wave).
__global__ void gcn_scatter64(const int* __restrict__ src, const int* __restrict__ dst,
                              const float* __restrict__ dinv, const float* __restrict__ tmp,
                              float* __restrict__ out, int E) {
    long tid = (long)blockIdx.x * blockDim.x + threadIdx.x;
    int e = (int)(tid >> 4);
    if (e >= E) return;
    int lane = (int)(tid & 15);
    int s = src[e], d = dst[e];
    float nrm = dinv[s] * dinv[d];
    const float4 v = *(const float4*)(tmp + (size_t)s * 64 + lane * 4);
    float* o = out + (size_t)d * 64 + lane * 4;
    atomicAdd(o + 0, v.x * nrm);
    atomicAdd(o + 1, v.y * nrm);
    atomicAdd(o + 2, v.z * nrm);
    atomicAdd(o + 3, v.w * nrm);
}

// ---------------- bias (+ optional relu), float4 elementwise ----------------
template<int D, bool RELU>
__global__ void gcn_bias_act(float* __restrict__ h, const float* __restrict__ b, long n4) {
    long i = (long)blockIdx.x * blockDim.x + threadIdx.x;
    if (i >= n4) return;
    int c4 = (int)(i & (D / 4 - 1));
    float4 bb = ((const float4*)b)[c4];
    float4 v  = ((float4*)h)[i];
    v.x += bb.x; v.y += bb.y; v.z += bb.z; v.w += bb.w;
    if (RELU) {
        v.x = fmaxf(v.x, 0.f); v.y = fmaxf(v.y, 0.f);
        v.z = fmaxf(v.z, 0.f); v.w = fmaxf(v.w, 0.f);
    }
    ((float4*)h)[i] = v;
}

// ---------------- driver ----------------

extern "C" void kernel_launch(void* const* d_in, const int* in_sizes, int n_in,
                              void* d_out, int out_size, void* d_ws, size_t ws_size,
                              hipStream_t stream) {
    const float* x  = (const float*)d_in[0];
    const int*   ei = (const int*)d_in[1];
    const float* W1 = (const float*)d_in[2];
    const float* b1 = (const float*)d_in[3];
    const float* W2 = (const float*)d_in[4];
    const float* b2 = (const float*)d_in[5];
    const float* W3 = (const float*)d_in[6];
    const float* b3 = (const float*)d_in[7];
    const float* W4 = (const float*)d_in[8];
    const float* b4 = (const float*)d_in[9];

    const int Nn = in_sizes[0] / DINK;   // 100000 (divisible by 16)
    const int E  = in_sizes[1] / 2;      // 1600000
    const int* srcI = ei;
    const int* dstI = ei + E;

    float* ws_f = (float*)d_ws;
    float* dinv = ws_f;
    size_t offA = ((size_t)Nn + 1023) & ~(size_t)1023;
    float* A = ws_f + offA;                 // GEMM output  [Nn x 128]
    float* B = A + (size_t)Nn * DHK;        // aggregation  [Nn x 128]
    float* out = (float*)d_out;             // [Nn x 64]

    const int TB = 256;
    const long nh4   = (long)Nn * DHK / 4;  // float4 count of a [Nn x 128] buffer
    const long no4   = (long)Nn * DOK / 4;  // float4 count of output
    const int  gN16  = Nn / 16;             // 6250 M-tiles
    const int  gSc128 = (int)(((long)E * 32 + TB - 1) / TB);
    const int  gSc64  = (int)(((long)E * 16 + TB - 1) / TB);
    const int  gH4    = (int)((nh4 + TB - 1) / TB);
    const int  gO4    = (int)((no4 + TB - 1) / TB);

    // degree -> dinv (in-place)
    gcn_zero_f4<<<(Nn / 4 + TB - 1) / TB, TB, 0, stream>>>((float4*)dinv, Nn / 4);
    gcn_deg<<<(E + TB - 1) / TB, TB, 0, stream>>>(dstI, dinv, E);
    gcn_dinv<<<(Nn + TB - 1) / TB, TB, 0, stream>>>(dinv, Nn);

    // Layer 1: x -> A -> B
    gcn_gemm_wmma<8><<<gN16, 256, 0, stream>>>(x, W1, A);
    gcn_zero_f4<<<gH4, TB, 0, stream>>>((float4*)B, nh4);
    gcn_scatter128<<<gSc128, TB, 0, stream>>>(srcI, dstI, dinv, A, B, E);
    gcn_bias_act<128, true><<<gH4, TB, 0, stream>>>(B, b1, nh4);

    // Layer 2: B -> A -> B
    gcn_gemm_wmma<8><<<gN16, 256, 0, stream>>>(B, W2, A);
    gcn_zero_f4<<<gH4, TB, 0, stream>>>((float4*)B, nh4);
    gcn_scatter128<<<gSc128, TB, 0, stream>>>(srcI, dstI, dinv, A, B, E);
    gcn_bias_act<128, true><<<gH4, TB, 0, stream>>>(B, b2, nh4);

    // Layer 3: B -> A -> B
    gcn_gemm_wmma<8><<<gN16, 256, 0, stream>>>(B, W3, A);
    gcn_zero_f4<<<gH4, TB, 0, stream>>>((float4*)B, nh4);
    gcn_scatter128<<<gSc128, TB, 0, stream>>>(srcI, dstI, dinv, A, B, E);
    gcn_bias_act<128, true><<<gH4, TB, 0, stream>>>(B, b3, nh4);

    // Layer 4: B -> A[Nn x 64] -> d_out (bias, no relu)
    gcn_gemm_wmma<4><<<gN16, 128, 0, stream>>>(B, W4, A);
    gcn_zero_f4<<<gO4, TB, 0, stream>>>((float4*)out, no4);
    gcn_scatter64<<<gSc64, TB, 0, stream>>>(srcI, dstI, dinv, A, out, E);
    gcn_bias_act<64, false><<<gO4, TB, 0, stream>>>(out, b4, no4);
}